// CODAPromptPool_8169027797033
// MI455X (gfx1250) — compile-verified
//
#include <hip/hip_runtime.h>

typedef __attribute__((ext_vector_type(2))) float v2f;
typedef __attribute__((ext_vector_type(4))) float v4f;
typedef __attribute__((ext_vector_type(8))) float v8f;

#define B_      64
#define S_      2048
#define D_      768
#define POOL_   32
#define L_      8
#define TOPK_   4
#define SCHUNKS 16
#define SROWS   (S_ / SCHUNKS)          /* 128 rows per chunk   */
#define PRE     (L_ + TOPK_ * L_ + 1)   /* 41 prepended rows    */
#define OUTROWS (PRE + S_)              /* 2089                 */
#define D4      (D_ / 4)                /* 192 float4 per row   */

// ---------------------------------------------------------------------------
// Kernel 1: stream x -> out (rows offset by PRE) and accumulate per-chunk
// partial sums for the mean-query. Non-temporal B128 ops: data has no reuse
// (403 MB > 192 MB L2), so bypass-hint both directions.
// Grid: 64*16 blocks, 192 threads (one v4f column per thread).
// ---------------------------------------------------------------------------
__global__ void coda_copy_reduce(const float* __restrict__ x,
                                 float* __restrict__ out,
                                 float* __restrict__ partials) {
    const int blk = blockIdx.x;
    const int b  = blk >> 4;      // batch
    const int sc = blk & 15;      // s-chunk
    const int t  = threadIdx.x;   // 0..191

    const v4f* src = (const v4f*)(x + ((size_t)b * S_ + (size_t)sc * SROWS) * D_) + t;
    v4f*       dst = (v4f*)(out + ((size_t)b * OUTROWS + PRE + (size_t)sc * SROWS) * D_) + t;

    v4f acc = {0.f, 0.f, 0.f, 0.f};
    #pragma unroll 4
    for (int s = 0; s < SROWS; ++s) {
        v4f v = __builtin_nontemporal_load(src);
        acc += v;
        __builtin_nontemporal_store(v, dst);
        src += D4; dst += D4;
    }
    v4f* p = (v4f*)(partials + ((size_t)(b * SCHUNKS + sc)) * D_) + t;
    *p = acc;
}

// ---------------------------------------------------------------------------
// Kernel 2: finalize queries (sum 16 partials, *1/S, l2-normalize) and
// l2-normalize keys. Blocks 0..63 = queries, 64..95 = keys. 256 threads,
// each owns 3 of the 768 dims.
// ---------------------------------------------------------------------------
__global__ void coda_normalize(const float* __restrict__ partials,
                               const float* __restrict__ e_keys,
                               float* __restrict__ qn,
                               float* __restrict__ kn) {
    __shared__ float red[256];
    const int i = blockIdx.x;
    const int t = threadIdx.x;

    float v0, v1, v2;
    float* dst;
    if (i < B_) {
        const float* p = partials + (size_t)i * SCHUNKS * D_;
        v0 = v1 = v2 = 0.f;
        #pragma unroll
        for (int c = 0; c < SCHUNKS; ++c) {
            v0 += p[c * D_ + t];
            v1 += p[c * D_ + t + 256];
            v2 += p[c * D_ + t + 512];
        }
        const float invS = 1.0f / (float)S_;
        v0 *= invS; v1 *= invS; v2 *= invS;   // mean (exact, like reference)
        dst = qn + (size_t)i * D_;
    } else {
        const int kb = i - B_;
        const float* k = e_keys + (size_t)kb * D_;
        v0 = k[t]; v1 = k[t + 256]; v2 = k[t + 512];
        dst = kn + (size_t)kb * D_;
    }

    red[t] = v0 * v0 + v1 * v1 + v2 * v2;
    __syncthreads();
    for (int o = 128; o > 0; o >>= 1) {
        if (t < o) red[t] += red[t + o];
        __syncthreads();
    }
    const float inv = rsqrtf(red[0] + 1e-12f);
    dst[t]       = v0 * inv;
    dst[t + 256] = v1 * inv;
    dst[t + 512] = v2 * inv;
}

// ---------------------------------------------------------------------------
// Kernel 3: similarity GEMM [64x768]x[768x32]^T via V_WMMA_F32_16X16X4_F32,
// one 16x16 tile per wave (8 waves), K-loop 768/4 = 192 steps. Then top-4
// per row (first-index tie-break). Single workgroup; EXEC all-1s for WMMA.
// A frag (16x4 f32): lane r=lane&15 -> row M=m0+r; half=lane>>4 selects
// K pair {0,1} vs {2,3}. B frag mirrors the K pairing, N=n0+r per lane.
// C/D: vgpr v, lanes0-15 -> M=m0+v, lanes16-31 -> M=m0+v+8, N=n0+r.
// ---------------------------------------------------------------------------
__global__ void coda_sim_topk(const float* __restrict__ qn,
                              const float* __restrict__ kn,
                              int* __restrict__ topk) {
    __shared__ float sim[B_][POOL_];

    const int t    = threadIdx.x;
    const int w    = t >> 5;
    const int lane = t & 31;
    const int m0   = (w & 3) << 4;   // 0,16,32,48
    const int n0   = (w >> 2) << 4;  // 0,16
    const int half = lane >> 4;
    const int r    = lane & 15;
    const int ka   = half ? 2 : 0;
    const int kb   = half ? 3 : 1;

    const float* qrow = qn + (size_t)(m0 + r) * D_;
    const float* krow = kn + (size_t)(n0 + r) * D_;

    v8f c = {0.f, 0.f, 0.f, 0.f, 0.f, 0.f, 0.f, 0.f};
    for (int k0 = 0; k0 < D_; k0 += 4) {
        v2f a, b;
        a.x = qrow[k0 + ka]; a.y = qrow[k0 + kb];
        b.x = krow[k0 + ka]; b.y = krow[k0 + kb];
        c = __builtin_amdgcn_wmma_f32_16x16x4_f32(
                /*neg_a=*/false, a, /*neg_b=*/false, b,
                /*c_mod=*/(short)0, c, /*reuse_a=*/false, /*reuse_b=*/false);
    }
    #pragma unroll
    for (int v = 0; v < 8; ++v)
        sim[m0 + v + 8 * half][n0 + r] = c[v];
    __syncthreads();

    if (t < B_) {
        float row[POOL_];
        #pragma unroll
        for (int j = 0; j < POOL_; ++j) row[j] = sim[t][j];
        #pragma unroll
        for (int sel = 0; sel < TOPK_; ++sel) {
            int best = 0; float bv = row[0];
            for (int j = 1; j < POOL_; ++j)
                if (row[j] > bv) { bv = row[j]; best = j; }   // strict > : first index wins ties
            topk[t * TOPK_ + sel] = best;
            row[best] = -__builtin_huge_valf();
        }
    }
}

// ---------------------------------------------------------------------------
// Kernel 4: write the 41 prepended rows per batch:
// rows 0..7 = g_prompts[task_id], 8..39 = gathered e_prompts, 40 = cls.
// ---------------------------------------------------------------------------
__global__ void coda_assemble(const float* __restrict__ g_prompts,
                              const float* __restrict__ e_prompts,
                              const float* __restrict__ cls_token,
                              const int* __restrict__ task_id,
                              const int* __restrict__ topk,
                              float* __restrict__ out) {
    const int b = blockIdx.x;
    const int t = threadIdx.x;
    __shared__ int idx[TOPK_];
    __shared__ int task;
    if (t == 0) task = *task_id;
    if (t < TOPK_) idx[t] = topk[b * TOPK_ + t];
    __syncthreads();

    v4f* dst = (v4f*)(out + (size_t)b * OUTROWS * D_);
    for (int i = t; i < PRE * D4; i += blockDim.x) {
        const int row = i / D4;
        const int c4  = i % D4;
        const v4f* src;
        if (row < L_) {
            src = (const v4f*)(g_prompts + ((size_t)task * L_ + row) * D_) + c4;
        } else if (row < L_ + TOPK_ * L_) {
            const int e = row - L_;
            const int eb = idx[e >> 3];           // which pool entry
            src = (const v4f*)(e_prompts + ((size_t)eb * L_ + (e & 7)) * D_) + c4;
        } else {
            src = (const v4f*)cls_token + c4;
        }
        dst[i] = *src;
    }
}

// ---------------------------------------------------------------------------
// Launch. ws layout (floats): [partials 64*16*768 | qn 64*768 | kn 32*768 |
// topk 64*4 ints]  ~= 3.44 MB total.
// ---------------------------------------------------------------------------
extern "C" void kernel_launch(void* const* d_in, const int* in_sizes, int n_in,
                              void* d_out, int out_size, void* d_ws, size_t ws_size,
                              hipStream_t stream) {
    const float* x         = (const float*)d_in[0];
    const float* g_prompts = (const float*)d_in[1];
    const float* e_prompts = (const float*)d_in[2];
    const float* e_keys    = (const float*)d_in[3];
    const float* cls_token = (const float*)d_in[4];
    const int*   task_id   = (const int*)d_in[5];
    float* out = (float*)d_out;

    float* ws       = (float*)d_ws;
    float* partials = ws;                                   // 786432 floats
    float* qn       = partials + (size_t)B_ * SCHUNKS * D_; //  49152 floats
    float* kn       = qn + (size_t)B_ * D_;                 //  24576 floats
    int*   topk     = (int*)(kn + (size_t)POOL_ * D_);      //    256 ints

    coda_copy_reduce<<<dim3(B_ * SCHUNKS), dim3(192), 0, stream>>>(x, out, partials);
    coda_normalize  <<<dim3(B_ + POOL_),   dim3(256), 0, stream>>>(partials, e_keys, qn, kn);
    coda_sim_topk   <<<dim3(1),            dim3(256), 0, stream>>>(qn, kn, topk);
    coda_assemble   <<<dim3(B_),           dim3(256), 0, stream>>>(g_prompts, e_prompts,
                                                                   cls_token, task_id, topk, out);
}